// SGC_35914516529298
// MI455X (gfx1250) — compile-verified
//
#include <hip/hip_runtime.h>
#include <hip/hip_bf16.h>

typedef __attribute__((ext_vector_type(2))) float v2f;
typedef __attribute__((ext_vector_type(8))) float v8f;

// ---------------------------------------------------------------------------
// Zero a float buffer (offsets here are 16B-aligned, sizes multiple of 4).
// ---------------------------------------------------------------------------
__global__ void sgc_zero_f32(float* __restrict__ p, long n) {
    long i = ((long)blockIdx.x * blockDim.x + threadIdx.x) * 4;
    if (i + 3 < n) {
        float4 z = {0.f, 0.f, 0.f, 0.f};
        *(float4*)(p + i) = z;
    } else {
        for (long j = i; j < n; ++j) p[j] = 0.f;
    }
}

// ---------------------------------------------------------------------------
// Degree accumulation: deg_out[src[e]] += 1, deg_in[dst[e]] += 1.
// Counts < 2^24 so float accumulation is exact.
// ---------------------------------------------------------------------------
__global__ void sgc_degrees(const int* __restrict__ src, const int* __restrict__ dst,
                            float* __restrict__ deg_out, float* __restrict__ deg_in, int E) {
    int e = blockIdx.x * blockDim.x + threadIdx.x;
    if (e < E) {
        atomicAdd(&deg_out[src[e]], 1.0f);
        atomicAdd(&deg_in[dst[e]], 1.0f);
    }
}

// ---------------------------------------------------------------------------
// Per-node norms; scales fused for the two hops + final GEMM:
//   s1 = norm_out, s2 = norm_in*norm_out, s3 = norm_in
// ---------------------------------------------------------------------------
__global__ void sgc_norms(const float* __restrict__ deg_out, const float* __restrict__ deg_in,
                          float* __restrict__ s1, float* __restrict__ s2,
                          float* __restrict__ s3, int N) {
    int i = blockIdx.x * blockDim.x + threadIdx.x;
    if (i < N) {
        float dout = deg_out[i];
        float din  = deg_in[i];
        float no = dout > 0.f ? rsqrtf(fmaxf(dout, 1.0f)) : 0.0f;
        float ni = din  > 0.f ? rsqrtf(fmaxf(din,  1.0f)) : 0.0f;
        s1[i] = no;
        s2[i] = ni * no;
        s3[i] = ni;
    }
}

// ---------------------------------------------------------------------------
// Edge scatter: out[dst[e]][:] += h[src[e]][:] * scale[src[e]]
// 16 lanes per edge, float4 gather per lane, 4 float atomics per lane.
// h (25.6MB) is resident in the 192MB L2, so gathers + atomics stay on-chip.
// ---------------------------------------------------------------------------
__global__ void __launch_bounds__(256) sgc_scatter(
        const float* __restrict__ h, const float* __restrict__ scale,
        const int* __restrict__ src, const int* __restrict__ dst,
        float* __restrict__ out, int E) {
    int gid = blockIdx.x * 256 + threadIdx.x;
    int e  = gid >> 4;
    int fg = (gid & 15) * 4;
    if (e >= E) return;
    int s = src[e];
    int d = dst[e];
    float sc = scale[s];
    float4 v = *(const float4*)(h + (long)s * 64 + fg);
    float* o = out + (long)d * 64 + fg;
    atomicAdd(o + 0, v.x * sc);
    atomicAdd(o + 1, v.y * sc);
    atomicAdd(o + 2, v.z * sc);
    atomicAdd(o + 3, v.w * sc);
}

// ---------------------------------------------------------------------------
// Final projection: out = relu( (hout * s3[:,None]) @ W ), fp32 WMMA, in place.
// V_WMMA_F32_16X16X4_F32: D(16x16) += A(16x4) x B(4x16), K stepped by 4.
// Fragment layouts (ISA 7.12.2):
//   A: lanes 0-15 hold K={k0,k0+1} of row m0+lane; lanes 16-31 hold K={k0+2,k0+3}
//   B: lanes 0-15 hold rows K={k0,k0+1} at N=lane%16; lanes 16-31 rows K={k0+2,k0+3}
//   C/D: VGPR j = row m0 + 8*(lane>=16) + j, col = lane%16
//
// W is staged in LDS pair-interleaved + bank-skewed:
//   lds[(k>>1)*PSTR + n*2 + (k&1)] = W[k][n],  PSTR = 160 dwords
// so each B fragment is ONE aligned ds_load_b64; PSTR % 64 == 32 puts the two
// lane halves (adjacent K-pairs) on disjoint bank halves -> conflict-free.
// ---------------------------------------------------------------------------
#define PSTR 160

__global__ void __launch_bounds__(256) sgc_gemm_relu(
        float* __restrict__ hout,       // [N,64] in: aggregated h, out: relu(h@W)
        const float* __restrict__ W,    // [64,64] row-major W[k][n]
        const float* __restrict__ s3,   // norm_in per row (fused into A load)
        int N) {
    __shared__ float ldsW[32 * PSTR];   // 20KB
    for (int i = threadIdx.x; i < 64 * 64; i += 256) {
        int k = i >> 6;
        int n = i & 63;
        ldsW[(k >> 1) * PSTR + (n << 1) + (k & 1)] = W[i];
    }
    __syncthreads();

    const int lane = threadIdx.x & 31;
    const int wv   = threadIdx.x >> 5;
    const int half = lane >> 4;          // 0 -> K pair {0,1}, 1 -> K pair {2,3}
    const int lr   = lane & 15;
    const int m0   = blockIdx.x * 128 + wv * 16;

    const int  rowA  = m0 + lr;
    const bool valid = rowA < N;
    const float sc   = valid ? s3[rowA] : 0.0f;     // invalid rows contribute 0
    const float* arow = hout + (long)(valid ? rowA : 0) * 64;

    v8f acc0 = {}; v8f acc1 = {}; v8f acc2 = {}; v8f acc3 = {};

    for (int kk = 0; kk < 16; ++kk) {
        const int k0 = kk * 4 + half * 2;           // even
        v2f a;
        a.x = arow[k0 + 0] * sc;
        a.y = arow[k0 + 1] * sc;

        const float* bp = ldsW + (k0 >> 1) * PSTR + (lr << 1);
        v2f b0 = *(const v2f*)(bp +  0);            // n-tile 0: cols  0..15
        v2f b1 = *(const v2f*)(bp + 32);            // n-tile 1: cols 16..31
        v2f b2 = *(const v2f*)(bp + 64);            // n-tile 2: cols 32..47
        v2f b3 = *(const v2f*)(bp + 96);            // n-tile 3: cols 48..63

        acc0 = __builtin_amdgcn_wmma_f32_16x16x4_f32(false, a, false, b0, (short)0, acc0, false, false);
        acc1 = __builtin_amdgcn_wmma_f32_16x16x4_f32(false, a, false, b1, (short)0, acc1, false, false);
        acc2 = __builtin_amdgcn_wmma_f32_16x16x4_f32(false, a, false, b2, (short)0, acc2, false, false);
        acc3 = __builtin_amdgcn_wmma_f32_16x16x4_f32(false, a, false, b3, (short)0, acc3, false, false);
    }

    // All A reads for this wave's 16-row band are done; safe to store in place.
    const int rbase = m0 + half * 8;
    for (int j = 0; j < 8; ++j) {
        int row = rbase + j;
        if (row < N) {
            float* orow = hout + (long)row * 64;
            orow[ 0 + lr] = fmaxf(acc0[j], 0.0f);
            orow[16 + lr] = fmaxf(acc1[j], 0.0f);
            orow[32 + lr] = fmaxf(acc2[j], 0.0f);
            orow[48 + lr] = fmaxf(acc3[j], 0.0f);
        }
    }
}

// ---------------------------------------------------------------------------
extern "C" void kernel_launch(void* const* d_in, const int* in_sizes, int n_in,
                              void* d_out, int out_size, void* d_ws, size_t ws_size,
                              hipStream_t stream) {
    const float* x   = (const float*)d_in[0];   // [N,64]
    const float* W   = (const float*)d_in[1];   // [64,64]
    const int*   src = (const int*)d_in[2];     // [E]
    const int*   dst = (const int*)d_in[3];     // [E]
    const int N = in_sizes[0] / 64;
    const int E = in_sizes[2];
    float* out = (float*)d_out;                 // [N,64]

    // Workspace layout (floats): deg_out[N] deg_in[N] s1[N] s2[N] s3[N] t1[N*64]
    float* ws      = (float*)d_ws;
    float* deg_out = ws;
    float* deg_in  = ws + 1L * N;
    float* s1      = ws + 2L * N;
    float* s2      = ws + 3L * N;
    float* s3      = ws + 4L * N;
    float* t1      = ws + 5L * N;

    const long zdeg = 2L * N;
    const long zh   = 64L * N;
    sgc_zero_f32<<<(int)((zdeg / 4 + 255) / 256), 256, 0, stream>>>(deg_out, zdeg);
    sgc_zero_f32<<<(int)((zh / 4 + 255) / 256), 256, 0, stream>>>(t1, zh);
    sgc_zero_f32<<<(int)((zh / 4 + 255) / 256), 256, 0, stream>>>(out, zh);

    sgc_degrees<<<(E + 255) / 256, 256, 0, stream>>>(src, dst, deg_out, deg_in, E);
    sgc_norms<<<(N + 255) / 256, 256, 0, stream>>>(deg_out, deg_in, s1, s2, s3, N);

    // Hop 1: t1[dst] += x[src] * norm_out[src]
    int sblocks = (int)(((long)E * 16 + 255) / 256);
    sgc_scatter<<<sblocks, 256, 0, stream>>>(x, s1, src, dst, t1, E);
    // Hop 2: out[dst] += t1[src] * (norm_in*norm_out)[src]
    sgc_scatter<<<sblocks, 256, 0, stream>>>(t1, s2, src, dst, out, E);

    // Final: out = relu((out * norm_in[:,None]) @ W), in place via f32 WMMA
    sgc_gemm_relu<<<(N + 127) / 128, 256, 0, stream>>>(out, W, s3, N);
}